// OptimizedMultiScaleCAB2_35845797052912
// MI455X (gfx1250) — compile-verified
//
#include <hip/hip_runtime.h>

typedef __attribute__((ext_vector_type(2))) float v2f;
typedef __attribute__((ext_vector_type(8))) float v8f;

#define BATCH 32
#define CH    384
#define HW    4096
#define BC    (BATCH * CH)          // 12288 (b,c) planes
#define NCOMB 24                    // 3*R concatenated features

// Workspace layout (floats):
//   [0,      BC)        avg pool
//   [BC,     2*BC)      max pool
//   [2*BC,   3*BC)      min pool
//   [3*BC,   3*BC+768)  combined [32][24]
//   [3*BC+768, +BC)     logits   [32][384]
#define WS_COMB   (3 * BC)
#define WS_LOGITS (3 * BC + BATCH * NCOMB)

// ---------------------------------------------------------------------------
// K1: fused global avg/max/min pool. One block per (b,c) plane of 4096 f32.
// 256 threads x 4 float4 loads each = 4096 elements, fully coalesced.
// ---------------------------------------------------------------------------
__global__ __launch_bounds__(256)
void pool_kernel(const float* __restrict__ x, float* __restrict__ ws) {
    __shared__ float ssum[256], smax[256], smin[256];
    const int bc = blockIdx.x;
    const int t  = threadIdx.x;
    const float4* px = (const float4*)(x + (size_t)bc * HW);

    float s = 0.0f, mx = -INFINITY, mn = INFINITY;
#pragma unroll
    for (int r = 0; r < 4; ++r) {
        float4 v = px[t + 256 * r];
        s  += (v.x + v.y) + (v.z + v.w);
        mx = fmaxf(mx, fmaxf(fmaxf(v.x, v.y), fmaxf(v.z, v.w)));
        mn = fminf(mn, fminf(fminf(v.x, v.y), fminf(v.z, v.w)));
    }
    ssum[t] = s; smax[t] = mx; smin[t] = mn;
    __syncthreads();
#pragma unroll
    for (int stride = 128; stride > 0; stride >>= 1) {
        if (t < stride) {
            ssum[t] += ssum[t + stride];
            smax[t] = fmaxf(smax[t], smax[t + stride]);
            smin[t] = fminf(smin[t], smin[t + stride]);
        }
        __syncthreads();
    }
    if (t == 0) {
        ws[bc]          = ssum[0] * (1.0f / HW);   // mean
        ws[BC + bc]     = smax[0];
        ws[2 * BC + bc] = smin[0];
    }
}

// ---------------------------------------------------------------------------
// K2: build combined[32][24] = [avg@W1.T | grouped(max) | min@W3.T].
// 768 dot products of length <= 384: trivial, one 256-thread block.
// ---------------------------------------------------------------------------
__global__ __launch_bounds__(256)
void combined_kernel(const float* __restrict__ W1, const float* __restrict__ W2_1,
                     const float* __restrict__ W2_2, const float* __restrict__ W3,
                     float* __restrict__ ws) {
    const float* avg = ws;
    const float* mxp = ws + BC;
    const float* mnp = ws + 2 * BC;
    float* comb = ws + WS_COMB;

    for (int task = threadIdx.x; task < BATCH * NCOMB; task += 256) {
        const int b = task / NCOMB;
        const int j = task % NCOMB;
        float acc = 0.0f;
        if (j < 8) {                       // avg_out = avg @ W1.T
            for (int c = 0; c < CH; ++c) acc += avg[b * CH + c] * W1[j * CH + c];
        } else if (j < 16) {               // grouped conv on max pool
            const int r = j - 8;
            for (int g = 0; g < 48; ++g) acc += mxp[b * CH + r * 48 + g] * W2_1[r * 48 + g];
            acc *= W2_2[r];
        } else {                           // min_out = min @ W3.T
            const int r = j - 16;
            for (int c = 0; c < CH; ++c) acc += mnp[b * CH + c] * W3[r * CH + c];
        }
        comb[b * NCOMB + j] = acc;
    }
}

// ---------------------------------------------------------------------------
// K3: logits[32][384] = combined[32][24] @ Wfc.T[24][384] via
// V_WMMA_F32_16X16X4_F32. 24 blocks (one 16-col tile each) x 1 wave32.
// M=32 -> two 16-row tiles, K=24 -> six K=4 steps => 12 WMMAs per wave.
// EXEC all-ones: 32 threads, no divergence.
// ---------------------------------------------------------------------------
__global__ __launch_bounds__(32)
void logits_wmma_kernel(const float* __restrict__ Wfc, float* __restrict__ ws) {
    const float* comb   = ws + WS_COMB;
    float*       logits = ws + WS_LOGITS;

    const int lane  = threadIdx.x;        // 0..31
    const int cbase = blockIdx.x * 16;    // N tile base (column in C)
    const int col   = lane & 15;          // N within tile / M row for A
    const int koff  = (lane >> 4) << 1;   // 0 for lanes 0-15, 2 for lanes 16-31

    // B fragments: B[K][N] = Wfc[N][K]  (Wfc is [C][24] row-major => K contiguous)
    v2f bfrag[6];
#pragma unroll
    for (int kk = 0; kk < 6; ++kk) {
        const float* p = Wfc + (size_t)(cbase + col) * NCOMB + kk * 4 + koff;
        bfrag[kk] = *(const v2f*)p;       // {B[kbase+koff][col], B[kbase+koff+1][col]}
    }

#pragma unroll
    for (int mtile = 0; mtile < 2; ++mtile) {
        v8f acc = {};
#pragma unroll
        for (int kk = 0; kk < 6; ++kk) {
            // A fragment: row = mtile*16 + (lane&15) in BOTH half-waves, K split by koff
            const float* pa = comb + (size_t)(mtile * 16 + col) * NCOMB + kk * 4 + koff;
            v2f afrag = *(const v2f*)pa;
            acc = __builtin_amdgcn_wmma_f32_16x16x4_f32(
                /*neg_a=*/false, afrag, /*neg_b=*/false, bfrag[kk],
                /*c_mod=*/(short)0, acc, /*reuse_a=*/false, /*reuse_b=*/false);
        }
        // D layout: VGPR i -> M=i (lanes 0-15), M=8+i (lanes 16-31)
        const int rbase = mtile * 16 + ((lane >> 4) << 3);
#pragma unroll
        for (int i = 0; i < 8; ++i)
            logits[(size_t)(rbase + i) * CH + cbase + col] = acc[i];
    }
}

// ---------------------------------------------------------------------------
// K4: out = sigmoid(x + logits[b,c]) elementwise, float4 vectorized.
// 1024 float4 per (b,c) plane -> plane index = i4 >> 10.
// ---------------------------------------------------------------------------
__global__ __launch_bounds__(256)
void sigmoid_kernel(const float* __restrict__ x, const float* __restrict__ logits,
                    float* __restrict__ out) {
    const size_t i4 = (size_t)blockIdx.x * 256 + threadIdx.x;
    const float4* px = (const float4*)x;
    float4*       po = (float4*)out;

    const int bc = (int)(i4 >> 10);
    const float l = logits[bc];
    float4 v = px[i4];
    float4 o;
    o.x = 1.0f / (1.0f + __expf(-(v.x + l)));
    o.y = 1.0f / (1.0f + __expf(-(v.y + l)));
    o.z = 1.0f / (1.0f + __expf(-(v.z + l)));
    o.w = 1.0f / (1.0f + __expf(-(v.w + l)));
    po[i4] = o;
}

extern "C" void kernel_launch(void* const* d_in, const int* in_sizes, int n_in,
                              void* d_out, int out_size, void* d_ws, size_t ws_size,
                              hipStream_t stream) {
    const float* x    = (const float*)d_in[0];
    const float* W1   = (const float*)d_in[1];
    const float* W2_1 = (const float*)d_in[2];
    const float* W2_2 = (const float*)d_in[3];
    const float* W3   = (const float*)d_in[4];
    const float* Wfc  = (const float*)d_in[5];
    float* ws  = (float*)d_ws;
    float* out = (float*)d_out;

    pool_kernel<<<BC, 256, 0, stream>>>(x, ws);
    combined_kernel<<<1, 256, 0, stream>>>(W1, W2_1, W2_2, W3, ws);
    logits_wmma_kernel<<<CH / 16, 32, 0, stream>>>(Wfc, ws);
    sigmoid_kernel<<<(BC * HW) / (4 * 256), 256, 0, stream>>>(x, ws + WS_LOGITS, out);
}